// Tagger_13340168421424
// MI455X (gfx1250) — compile-verified
//
#include <hip/hip_runtime.h>

typedef __attribute__((ext_vector_type(2))) float v2f;
typedef __attribute__((ext_vector_type(8))) float v8f;

namespace {
constexpr int WIN     = 5;
constexpr int VOCAB   = 100000;
constexpr int EMB     = 50;
constexpr int HID     = 250;   // == WIN*EMB
constexpr int OUTD    = 36;
constexpr int KPAD    = 256;   // padded K (and padded hidden width)
constexpr int NPAD2   = 48;    // padded output width (3 tiles of 16)
constexpr int ROWS    = 64;    // batch rows per workgroup (4 row tiles)
constexpr int STRIDE  = 260;   // LDS row stride in floats; 260 % 64 == 4 -> conflict-free
constexpr int THREADS = 128;   // 4 wave32
}

// Branch-free tanh: exact identity (e^2x - 1)/(e^2x + 1) on the TRANS pipe.
// Clamp keeps e^2x finite (avoids inf * rcp(inf) = NaN); |x|>=15 already
// rounds to +/-1 in f32. ~6 VALU/TRANS ops, no EXEC divergence.
__device__ __forceinline__ float fast_tanh(float x) {
  float xc = fminf(15.0f, fmaxf(-15.0f, x));
  float t  = __expf(2.0f * xc);                       // v_exp_f32
  return (t - 1.0f) * __builtin_amdgcn_rcpf(t + 1.0f); // v_rcp_f32
}

// ---- prep: zero-padded copies of the weights in workspace ----
// wp : [KPAD][KPAD]  = w_in  padded (k<250, n<250)
// wp2: [KPAD][NPAD2] = w_out padded (k<250, n<36)
__global__ __launch_bounds__(256)
void pad_weights(const float* __restrict__ w_in, const float* __restrict__ w_out,
                 float* __restrict__ wp, float* __restrict__ wp2)
{
  int i = blockIdx.x * 256 + threadIdx.x;
  if (i < KPAD * KPAD) {
    int k = i >> 8, n = i & 255;
    wp[i] = (k < HID && n < HID) ? w_in[k * HID + n] : 0.0f;
  }
  if (i < KPAD * NPAD2) {
    int k = i / NPAD2, n = i - k * NPAD2;
    wp2[i] = (k < HID && n < OUTD) ? w_out[k * OUTD + n] : 0.0f;
  }
}

__global__ __launch_bounds__(THREADS)
void tagger_fused(const int*   __restrict__ idx,
                  const float* __restrict__ emb,
                  const float* __restrict__ wp,    // padded w_in  [256][256]
                  const float* __restrict__ b_in,
                  const float* __restrict__ wp2,   // padded w_out [256][48]
                  const float* __restrict__ b_out,
                  float*       __restrict__ out)
{
  __shared__ float x_lds[ROWS * STRIDE];
  __shared__ float h_lds[ROWS * STRIDE];
  __shared__ int   i_lds[ROWS * WIN];

  const int t       = threadIdx.x;
  const int rowBase = blockIdx.x * ROWS;

  // ---- Phase 0: stage window indices, then gather embeddings into LDS ----
  for (int e = t; e < ROWS * WIN; e += THREADS)
    i_lds[e] = idx[rowBase * WIN + e];
  __syncthreads();

  for (int e = t; e < ROWS * HID; e += THREADS) {
    int row = e / HID;
    int col = e - row * HID;          // 0..249 (concat position)
    int w   = col / EMB;              // window slot
    int c   = col - w * EMB;          // dim within embedding
    int id  = i_lds[row * WIN + w];
    float v = 0.0f;
    if (id >= 0 && id < VOCAB) v = emb[id * EMB + c];   // coalesced along c
    x_lds[row * STRIDE + col] = v;
  }
  // zero K padding (cols 250..259) so padded WMMA K-steps contribute 0
  for (int p = t; p < ROWS * (STRIDE - HID); p += THREADS) {
    int row = p / (STRIDE - HID);
    int c   = HID + (p - row * (STRIDE - HID));
    x_lds[row * STRIDE + c] = 0.0f;
  }
  __syncthreads();

  const int lane = t & 31;
  const int wave = t >> 5;       // wave32: 4 waves
  const int hi   = lane >> 4;    // half-wave select (K pairing / M offset)
  const int nloc = lane & 15;    // N within tile (B/C fragments)
  const int m    = lane & 15;    // M within tile (A fragment)

  // ---- Phase 1: h = tanh(x @ w_in + b_in) ----
  // Each wave owns 4 column tiles (ct = 4p + wave); per k-step one B fragment
  // is shared by 4 WMMAs over the 4 row tiles -> 4 independent acc chains.
  const float* xr0 = &x_lds[( 0 + m) * STRIDE];
  const float* xr1 = &x_lds[(16 + m) * STRIDE];
  const float* xr2 = &x_lds[(32 + m) * STRIDE];
  const float* xr3 = &x_lds[(48 + m) * STRIDE];

  for (int p = 0; p < 4; ++p) {
    const int ct   = p * 4 + wave;
    const int coln = ct * 16 + nloc;                  // hidden column (0..255)
    const float bias = (coln < HID) ? b_in[coln] : 0.0f;
    v8f acc0 = {bias, bias, bias, bias, bias, bias, bias, bias};
    v8f acc1 = acc0, acc2 = acc0, acc3 = acc0;

    #pragma unroll 4
    for (int k = 0; k < KPAD; k += 4) {
      const int kk = k + 2 * hi;                      // K rows kk, kk+1
      v2f b;
      b.x = wp[kk * KPAD + coln];                     // coalesced row reads
      b.y = wp[(kk + 1) * KPAD + coln];
      v2f a0 = *reinterpret_cast<const v2f*>(&xr0[kk]);
      v2f a1 = *reinterpret_cast<const v2f*>(&xr1[kk]);
      v2f a2 = *reinterpret_cast<const v2f*>(&xr2[kk]);
      v2f a3 = *reinterpret_cast<const v2f*>(&xr3[kk]);
      acc0 = __builtin_amdgcn_wmma_f32_16x16x4_f32(false, a0, false, b, (short)0, acc0, false, false);
      acc1 = __builtin_amdgcn_wmma_f32_16x16x4_f32(false, a1, false, b, (short)0, acc1, false, false);
      acc2 = __builtin_amdgcn_wmma_f32_16x16x4_f32(false, a2, false, b, (short)0, acc2, false, false);
      acc3 = __builtin_amdgcn_wmma_f32_16x16x4_f32(false, a3, false, b, (short)0, acc3, false, false);
    }

    // padded columns are exactly 0 (bias 0, zero B) -> tanh(0)=0, store unmasked
    #pragma unroll
    for (int j = 0; j < 8; ++j) {                     // C: M = j + 8*hi, N = nloc
      const int M = j + 8 * hi;
      h_lds[( 0 + M) * STRIDE + coln] = fast_tanh(acc0[j]);
      h_lds[(16 + M) * STRIDE + coln] = fast_tanh(acc1[j]);
      h_lds[(32 + M) * STRIDE + coln] = fast_tanh(acc2[j]);
      h_lds[(48 + M) * STRIDE + coln] = fast_tanh(acc3[j]);
    }
  }
  __syncthreads();   // h written cross-wave (waves own columns, read rows)

  // ---- Phase 2: out = h @ w_out + b_out (3 column tiles cover OUT=36) ----
  const int rt = wave;                                // one row tile per wave
  const float* hrow = &h_lds[(rt * 16 + m) * STRIDE];

  for (int ct = 0; ct < 3; ++ct) {
    const int coln = ct * 16 + nloc;                  // output column (0..47)
    const float bias = (coln < OUTD) ? b_out[coln] : 0.0f;
    v8f acc = {bias, bias, bias, bias, bias, bias, bias, bias};

    #pragma unroll 4
    for (int k = 0; k < KPAD; k += 4) {
      const int kk = k + 2 * hi;
      v2f a = *reinterpret_cast<const v2f*>(&hrow[kk]);
      v2f b;
      b.x = wp2[kk * NPAD2 + coln];
      b.y = wp2[(kk + 1) * NPAD2 + coln];
      acc = __builtin_amdgcn_wmma_f32_16x16x4_f32(false, a, false, b, (short)0, acc, false, false);
    }

    #pragma unroll
    for (int j = 0; j < 8; ++j) {
      const int M = j + 8 * hi;
      if (coln < OUTD)
        out[(size_t)(rowBase + rt * 16 + M) * OUTD + coln] = acc[j];
    }
  }
}

extern "C" void kernel_launch(void* const* d_in, const int* in_sizes, int n_in,
                              void* d_out, int out_size, void* d_ws, size_t ws_size,
                              hipStream_t stream) {
  const int*   idx     = (const int*)  d_in[0];
  const float* emb     = (const float*)d_in[1];
  const float* w_in_p  = (const float*)d_in[2];
  const float* b_in_p  = (const float*)d_in[3];
  const float* w_out_p = (const float*)d_in[4];
  const float* b_out_p = (const float*)d_in[5];
  float* out = (float*)d_out;

  // workspace: padded weights (256*256 + 256*48 floats = ~311 KB)
  float* wp  = (float*)d_ws;
  float* wp2 = wp + KPAD * KPAD;

  pad_weights<<<(KPAD * KPAD + 255) / 256, 256, 0, stream>>>(w_in_p, w_out_p, wp, wp2);

  const int batch  = in_sizes[0] / WIN;   // 262144
  const int blocks = batch / ROWS;        // 4096 workgroups
  tagger_fused<<<blocks, THREADS, 0, stream>>>(idx, emb, wp, b_in_p, wp2, b_out_p, out);
}